// FlashAttentionPooling_32332513804506
// MI455X (gfx1250) — compile-verified
//
#include <hip/hip_runtime.h>
#include <hip/hip_bf16.h>
#include <math.h>

#define B_  16
#define T_  4096
#define D_  1024
#define S1_ 32            // segments for the mean pass
#define S_  64            // T-slices for the fused softmax pass
#define TSEG (T_ / S_)    // 64 rows per slice (256 KB -> L2 resident between phases)

typedef __attribute__((ext_vector_type(2))) float v2f;
typedef __attribute__((ext_vector_type(8))) float v8f;

union V8 { v8f v; float f[8]; };

// ---------------- pass 1: column mean of x over T ----------------

// part[b][s][d] = sum_{t in seg} x[b,t,d]
__global__ __launch_bounds__(256) void colsum_partial(const float* __restrict__ x,
                                                      float* __restrict__ part) {
    const int d = blockIdx.x * 256 + threadIdx.x;
    const int b = blockIdx.y;
    const int s = blockIdx.z;
    const int tseg = T_ / S1_;
    const float* p = x + ((size_t)b * T_ + (size_t)s * tseg) * D_ + d;
    float acc = 0.f;
    for (int t = 0; t < tseg; ++t) {
        __builtin_prefetch(p + 16 * D_, 0, 3);   // global_prefetch_b8
        acc += *p;
        p += D_;
    }
    part[((size_t)b * S1_ + s) * D_ + d] = acc;
}

// out[b][d] = scale * sum_s part[b][s][d]   (deterministic fixed-order reduce)
__global__ __launch_bounds__(256) void reduce_segments(const float* __restrict__ part,
                                                       float* __restrict__ out,
                                                       int nseg, float scale) {
    const int d = blockIdx.x * 256 + threadIdx.x;
    const int b = blockIdx.y;
    float acc = 0.f;
    for (int s = 0; s < nseg; ++s)
        acc += part[((size_t)b * nseg + s) * D_ + d];
    out[(size_t)b * D_ + d] = acc * scale;
}

// ---------------- WMMA 16x16 GEMM: C(16 x D) = A(16 x D) * B(D x D) * scale ----------------
// BT = true : B[k][n] = W[n*D + k]   (A @ W^T, used for q = xbar @ Wq^T)
// BT = false: B[k][n] = W[k*D + n]   (A @ W,   used for v = q @ Wk)
template <bool BT>
__global__ __launch_bounds__(32) void wmma_gemm16(const float* __restrict__ A,
                                                  const float* __restrict__ W,
                                                  float* __restrict__ C,
                                                  float scale) {
    const int lane = threadIdx.x;        // one full wave32, EXEC all ones
    const int half = lane >> 4;          // 0: lanes 0-15, 1: lanes 16-31
    const int mn   = lane & 15;          // M for A-frag, N for B-frag
    const int kb   = half * 2;           // K sub-offset within 16x4 / 4x16 frags
    const int n0   = blockIdx.x * 16;

    V8 acc = {};                         // 16x16 f32 accumulator (8 VGPRs)
    for (int k0 = 0; k0 < D_; k0 += 4) {
        const int k = k0 + kb;
        v2f a = { A[(size_t)mn * D_ + k], A[(size_t)mn * D_ + k + 1] };
        v2f b;
        if (BT) {
            b = v2f{ W[(size_t)(n0 + mn) * D_ + k], W[(size_t)(n0 + mn) * D_ + k + 1] };
        } else {
            b = v2f{ W[(size_t)k * D_ + n0 + mn], W[(size_t)(k + 1) * D_ + n0 + mn] };
        }
        acc.v = __builtin_amdgcn_wmma_f32_16x16x4_f32(
            false, a, false, b, (short)0, acc.v, false, false);
    }
#pragma unroll
    for (int r = 0; r < 8; ++r) {
        const int row = r + half * 8;    // C/D layout: VGPR r -> M = r + 8*half
        C[(size_t)row * D_ + n0 + mn] = acc.f[r] * scale;
    }
}

// ---------------- pass 2 (fused): scores + slice softmax + weighted accumulation ----------------
// Per (b, slice): phase A streams 64 rows from HBM for the dots; phase B re-reads
// them (L2-resident) for the weighted sum. Two-level softmax merged later.
__global__ __launch_bounds__(256) void fused_attn_partial(const float* __restrict__ x,
                                                          const float* __restrict__ v,
                                                          const unsigned char* __restrict__ mask,
                                                          float* __restrict__ ms,
                                                          float* __restrict__ ls,
                                                          float* __restrict__ acc) {
    __shared__ float p[TSEG];     // scores, then softmax weights
    __shared__ float red[256];
    const int s    = blockIdx.x;
    const int b    = blockIdx.y;
    const int tid  = threadIdx.x;
    const int wave = tid >> 5;
    const int lane = tid & 31;
    const int t0   = s * TSEG;
    const float4* vr = (const float4*)(v + (size_t)b * D_);

    // ---- phase A: dot(x[b,t,:], v[b,:]) per row, one row per wave per iter ----
    for (int i = wave; i < TSEG; i += 8) {
        const int t = t0 + i;
        const float4* xr = (const float4*)(x + ((size_t)b * T_ + t) * D_);
        __builtin_prefetch((const char*)xr + 8 * (size_t)D_ * 4, 0, 3);
        float sdot = 0.f;
#pragma unroll
        for (int j = 0; j < D_ / 4 / 32; ++j) {
            const int idx = j * 32 + lane;
            float4 a = xr[idx];
            float4 w = vr[idx];
            sdot = fmaf(a.x, w.x, sdot);
            sdot = fmaf(a.y, w.y, sdot);
            sdot = fmaf(a.z, w.z, sdot);
            sdot = fmaf(a.w, w.w, sdot);
        }
#pragma unroll
        for (int off = 16; off > 0; off >>= 1)
            sdot += __shfl_xor(sdot, off, 32);
        if (lane == 0)
            p[i] = mask[(size_t)b * T_ + t] ? sdot : -INFINITY;
    }
    __syncthreads();

    // ---- slice-local max ----
    float m = -INFINITY;
    for (int i = tid; i < TSEG; i += 256) m = fmaxf(m, p[i]);
    red[tid] = m; __syncthreads();
    for (int off = 128; off > 0; off >>= 1) {
        if (tid < off) red[tid] = fmaxf(red[tid], red[tid + off]);
        __syncthreads();
    }
    m = red[0];
    const float meff = (m == -INFINITY) ? 0.f : m;   // all-masked slice guard
    __syncthreads();

    // ---- weights + slice-local sum ----
    float l = 0.f;
    for (int i = tid; i < TSEG; i += 256) {
        const float w = __expf(p[i] - meff);          // exp(-inf) = 0 for masked
        p[i] = w;
        l += w;
    }
    red[tid] = l; __syncthreads();
    for (int off = 128; off > 0; off >>= 1) {
        if (tid < off) red[tid] += red[tid + off];
        __syncthreads();
    }
    if (tid == 0) {
        ms[(size_t)b * S_ + s] = meff;
        ls[(size_t)b * S_ + s] = red[0];
    }
    __syncthreads();   // all weights visible before phase B

    // ---- phase B: acc_s[d] = sum_t w[t] * x[b,t,d]  (rows hot in L2) ----
    float a0 = 0.f, a1 = 0.f, a2 = 0.f, a3 = 0.f;
    const float* base = x + ((size_t)b * T_ + t0) * D_;
    for (int i = 0; i < TSEG; ++i) {
        const float w   = p[i];
        const float* row = base + (size_t)i * D_;
        a0 = fmaf(w, row[tid],       a0);
        a1 = fmaf(w, row[tid + 256], a1);
        a2 = fmaf(w, row[tid + 512], a2);
        a3 = fmaf(w, row[tid + 768], a3);
    }
    float* ap = acc + ((size_t)b * S_ + s) * D_;
    ap[tid]       = a0;
    ap[tid + 256] = a1;
    ap[tid + 512] = a2;
    ap[tid + 768] = a3;
}

// ---------------- combine slices: exact two-level softmax merge ----------------
__global__ __launch_bounds__(256) void attn_combine(const float* __restrict__ ms,
                                                    const float* __restrict__ ls,
                                                    const float* __restrict__ acc,
                                                    float* __restrict__ out) {
    const int d = blockIdx.x * 256 + threadIdx.x;
    const int b = blockIdx.y;
    float m = -INFINITY;
    for (int s = 0; s < S_; ++s) m = fmaxf(m, ms[(size_t)b * S_ + s]);
    float L = 0.f;
    for (int s = 0; s < S_; ++s)
        L = fmaf(ls[(size_t)b * S_ + s], __expf(ms[(size_t)b * S_ + s] - m), L);
    float a = 0.f;
    for (int s = 0; s < S_; ++s)
        a = fmaf(acc[((size_t)b * S_ + s) * D_ + d],
                 __expf(ms[(size_t)b * S_ + s] - m), a);
    out[(size_t)b * D_ + d] = a / L;
}

// ---------------- host launcher ----------------
extern "C" void kernel_launch(void* const* d_in, const int* in_sizes, int n_in,
                              void* d_out, int out_size, void* d_ws, size_t ws_size,
                              hipStream_t stream) {
    (void)in_sizes; (void)n_in; (void)out_size; (void)ws_size;
    const float*         x    = (const float*)d_in[0];          // (B,T,D) f32
    const unsigned char* mask = (const unsigned char*)d_in[1];  // (B,T) bool
    const float*         Wq   = (const float*)d_in[2];          // (D,D) f32
    const float*         Wk   = (const float*)d_in[3];          // (D,D) f32
    float*               out  = (float*)d_out;                  // (B,D) f32

    float* ws   = (float*)d_ws;
    float* part = ws;                           // B*S1*D
    float* xbar = part + (size_t)B_ * S1_ * D_;
    float* q    = xbar + (size_t)B_ * D_;
    float* v    = q    + (size_t)B_ * D_;
    float* ms   = v    + (size_t)B_ * D_;       // B*S
    float* ls   = ms   + (size_t)B_ * S_;       // B*S
    float* acc  = ls   + (size_t)B_ * S_;       // B*S*D

    // 1. x̄[b] = mean_t x[b,t,:]            (HBM pass 1 over x)
    colsum_partial<<<dim3(D_ / 256, B_, S1_), 256, 0, stream>>>(x, part);
    reduce_segments<<<dim3(D_ / 256, B_), 256, 0, stream>>>(part, xbar, S1_, 1.0f / (float)T_);

    // 2. q = x̄ @ Wq^T                      (WMMA f32 16x16x4)
    wmma_gemm16<true><<<D_ / 16, 32, 0, stream>>>(xbar, Wq, q, 1.0f);

    // 3. v = (q @ Wk) / sqrt(D)
    wmma_gemm16<false><<<D_ / 16, 32, 0, stream>>>(q, Wk, v, 0.03125f);

    // 4. fused scores + slice softmax + weighted sum   (HBM pass 2 over x; re-read from L2)
    fused_attn_partial<<<dim3(S_, B_), 256, 0, stream>>>(x, v, mask, ms, ls, acc);

    // 5. merge slices
    attn_combine<<<dim3(D_ / 256, B_), 256, 0, stream>>>(ms, ls, acc, out);
}